// TrimZeros_58789512348330
// MI455X (gfx1250) — compile-verified
//
#include <hip/hip_runtime.h>
#include <stdint.h>

// TrimZeros for MI455X (gfx1250): pure bandwidth problem (zero matmul content,
// ~1.4 GiB of traffic -> ~60us at 23.3 TB/s). Uses wave32 shuffle reduction,
// b128 vector memory, NT cache hints, and the CDNA5 async global<->LDS path
// (ASYNCcnt) for the bulk streaming copy.

typedef float f32x4 __attribute__((ext_vector_type(4)));

#define TRIM_L 131072  // sequence length L from the reference (power of two)

// ---------------------------------------------------------------------------
// Workspace init: last-nonzero-column accumulator starts at -1 every call
// (graph-replay safe: no reliance on prior-call state).
// ---------------------------------------------------------------------------
__global__ void tz_init(int* __restrict__ last) { *last = -1; }

// ---------------------------------------------------------------------------
// Pass 1: global arg-reduction. For every nonzero element, its column index
// contributes to max. Per-thread max over grid-stride b128 loads, wave32
// shuffle tree, one atomicMax per wave.
// ---------------------------------------------------------------------------
__global__ __launch_bounds__(256) void tz_find_last(const f32x4* __restrict__ x,
                                                    int* __restrict__ last,
                                                    int n4) {
  int lmax = -1;
  const int stride = gridDim.x * blockDim.x;
  for (int i = blockIdx.x * blockDim.x + threadIdx.x; i < n4; i += stride) {
    // streaming read: non-temporal so we don't thrash L2 ahead of pass 2
    f32x4 v = __builtin_nontemporal_load(&x[i]);
    int j = (i << 2) & (TRIM_L - 1);  // column of element 0 of this float4
    int c = -1;
    if (v.x != 0.f) c = 0;
    if (v.y != 0.f) c = 1;
    if (v.z != 0.f) c = 2;
    if (v.w != 0.f) c = 3;
    if (c >= 0) lmax = max(lmax, j + c);
  }
  // wave32 reduction (MI455X is wave32-only)
  for (int off = 16; off > 0; off >>= 1)
    lmax = max(lmax, __shfl_down(lmax, off, 32));
  if ((threadIdx.x & 31) == 0 && lmax >= 0)
    atomicMax(last, lmax);
}

// ---------------------------------------------------------------------------
// Pass 2: masked streaming copy, one float4 per thread (exact grid).
//  - fully kept tile:    async copy Global -> LDS -> Global (ASYNCcnt path,
//                        no VGPR data movement)
//  - fully trimmed tile: NT store of zeros, NO load (saves tail-read BW)
//  - boundary tile:      per-element select (at most one float4 per row)
// ---------------------------------------------------------------------------
__global__ __launch_bounds__(256) void tz_mask_copy(const f32x4* __restrict__ x,
                                                    f32x4* __restrict__ out,
                                                    const int* __restrict__ last_p,
                                                    int n4) {
  __shared__ f32x4 stage[256];  // 16 B per lane staging buffer
  const int idx = blockIdx.x * blockDim.x + threadIdx.x;
  if (idx >= n4) return;

  const int last = __builtin_amdgcn_readfirstlane(*last_p);
  const int keep = (last < 0) ? TRIM_L : (last + 1);  // all-zero -> keep all
  const int j = (idx << 2) & (TRIM_L - 1);            // column of lane's elem 0

  const f32x4* src = x + idx;
  f32x4* dst = out + idx;

  if (j + 3 < keep) {
    // Bulk copy via CDNA5 async LDS ops. Low 32 bits of a generic shared
    // pointer are the workgroup-relative LDS byte offset (ISA 10.2 aperture
    // rules), which is what VDST/VSRC of the async ops expect.
    unsigned lds_off = (unsigned)(uintptr_t)&stage[threadIdx.x];
    asm volatile("global_load_async_to_lds_b128 %0, %1, off"
                 :
                 : "v"(lds_off), "v"(src)
                 : "memory");
    asm volatile("s_wait_asynccnt 0" ::: "memory");
    asm volatile("global_store_async_from_lds_b128 %0, %1, off"
                 :
                 : "v"(dst), "v"(lds_off)
                 : "memory");
    // S_ENDPGM performs an implicit wait-idle, so the outstanding async store
    // is guaranteed to drain before the wave retires.
  } else if (j >= keep) {
    f32x4 z = {0.f, 0.f, 0.f, 0.f};
    __builtin_nontemporal_store(z, dst);  // streaming store, no input read
  } else {
    f32x4 v = *src;
    v.x = (j + 0 < keep) ? v.x : 0.f;
    v.y = (j + 1 < keep) ? v.y : 0.f;
    v.z = (j + 2 < keep) ? v.z : 0.f;
    v.w = (j + 3 < keep) ? v.w : 0.f;
    *dst = v;
  }
}

// ---------------------------------------------------------------------------
// Host launcher
// ---------------------------------------------------------------------------
extern "C" void kernel_launch(void* const* d_in, const int* in_sizes, int n_in,
                              void* d_out, int out_size, void* d_ws, size_t ws_size,
                              hipStream_t stream) {
  const f32x4* x = (const f32x4*)d_in[0];
  f32x4* out = (f32x4*)d_out;
  int* d_last = (int*)d_ws;

  const int n = in_sizes[0];  // F * L = 134217728 for the reference shapes
  const int n4 = n >> 2;      // float4 tiles (L is a multiple of 4)

  tz_init<<<1, 1, 0, stream>>>(d_last);

  int red_blocks = (n4 + 255) / 256;
  if (red_blocks > 8192) red_blocks = 8192;  // grid-stride, ~16 tiles/thread
  tz_find_last<<<red_blocks, 256, 0, stream>>>(x, d_last, n4);

  const int copy_blocks = (n4 + 255) / 256;
  tz_mask_copy<<<copy_blocks, 256, 0, stream>>>(x, out, d_last, n4);
}